// WeightedDiceLoss_15702400434712
// MI455X (gfx1250) — compile-verified
//
#include <hip/hip_runtime.h>
#include <stdint.h>

typedef float v2f __attribute__((ext_vector_type(2)));
typedef float v8f __attribute__((ext_vector_type(8)));

#define IMG_H 512
#define IMG_W 512
#define TILE  64
#define HALO  15
#define RAD   30          // window size - 1
#define LT    94          // TILE + 2*HALO
#define LTS   96          // padded LDS stride for the raw tile
#define INV_KK (1.0f / 961.0f)
#define GAMMA 5.0f

// Zero the three f64 accumulators (inter, sum_iw, sum_tw) each call.
__global__ void wdice_init(double* acc) {
    if (threadIdx.x < 3) acc[threadIdx.x] = 0.0;
}

__global__ __launch_bounds__(256) void wdice_main(const float* __restrict__ inp,
                                                  const float* __restrict__ tgt,
                                                  double* __restrict__ acc) {
    __shared__ float sT[LT * LTS];     // raw target tile + halo   (~35.3 KB)
    __shared__ float sH[LT * TILE];    // horizontal box sums      (~24.1 KB)
    __shared__ float sRed[8 * 3];      // per-wave partials

    const int tid = threadIdx.x;
    const int x0 = blockIdx.x * TILE;
    const int y0 = blockIdx.y * TILE;
    const size_t base = (size_t)blockIdx.z * (IMG_H * IMG_W);
    const float* tbase = tgt + base;

    // ---- Stage 1: async-copy 94x94 target tile (zero-padded halo) into LDS ----
    // Border tiles: pre-zero LDS so EXEC-masked (out-of-image) lanes leave 0.
    if (x0 == 0 || y0 == 0 || x0 + TILE == IMG_W || y0 + TILE == IMG_H) {
        for (int e = tid; e < LT * LTS; e += 256) sT[e] = 0.0f;
        __syncthreads();   // order LDS zero-stores before async-engine writes
    }
    for (int e = tid; e < LT * LT; e += 256) {
        int r = e / LT;
        int c = e - r * LT;
        int y = y0 - HALO + r;
        int x = x0 - HALO + c;
        if ((unsigned)y < IMG_H && (unsigned)x < IMG_W) {
            // low 32 bits of a generic LDS pointer == wave-relative LDS byte addr
            unsigned lds_off = (unsigned)(uintptr_t)&sT[r * LTS + c];
            int voff = (y * IMG_W + x) * 4;
            // GVS mode: mem = SADDR_u64 + VADDR_i32 + inst_offset
            asm volatile("global_load_async_to_lds_b32 %0, %1, %2 offset:0"
                         :: "v"(lds_off), "v"(voff), "s"(tbase) : "memory");
        }
    }
    asm volatile("s_wait_asynccnt 0" ::: "memory");
    __syncthreads();

    // ---- Stage 2: horizontal 31-tap box sum, sliding window, 2 threads/row ----
    // sH[r][cx] = sum_{c=cx..cx+30} sT[r][c],  cx in [0,64)
    if (tid < 2 * LT) {
        const int r   = tid >> 1;
        const int cx0 = (tid & 1) * 32;
        float s = 0.0f;
        for (int c = cx0; c <= cx0 + RAD; ++c) s += sT[r * LTS + c];
        sH[r * TILE + cx0] = s;
        for (int cx = cx0 + 1; cx < cx0 + 32; ++cx) {
            s += sT[r * LTS + cx + RAD] - sT[r * LTS + cx - 1];
            sH[r * TILE + cx] = s;
        }
    }
    __syncthreads();

    // ---- Stage 3: vertical 31-tap box sum as banded f32 WMMA ----
    // OUT(16x16) = BAND(16x48) x Hsub(48x16), BAND[u][k] = (u <= k <= u+30)
    const int wave = tid >> 5;
    const int lane = tid & 31;
    const int ln   = lane & 15;
    const int hi   = lane >> 4;        // 0: lanes 0-15, 1: lanes 16-31

    // Band A depends only on (lane, step): hoist out of the subtile loop.
    // A layout (ISA 16x4 f32): lanes 0-15 hold K={4s,4s+1} for row M=ln;
    // lanes 16-31 hold K={4s+2,4s+3}.
    v2f Aband[12];
    #pragma unroll
    for (int s = 0; s < 12; ++s) {
        const int ka = 4 * s + 2 * hi;
        Aband[s].x = (ka     >= ln && ka     <= ln + RAD) ? 1.0f : 0.0f;
        Aband[s].y = (ka + 1 >= ln && ka + 1 <= ln + RAD) ? 1.0f : 0.0f;
    }

    float a0 = 0.0f, a1 = 0.0f, a2 = 0.0f;

    // 16 subtiles (4x4 grid of 16x16), 8 waves -> 2 subtiles per wave, uniform.
    for (int st = wave; st < 16; st += 8) {
        const int i = st >> 2;         // subtile row block
        const int j = st & 3;          // subtile col block
        v8f cacc = {};                 // C starts at zero

        #pragma unroll
        for (int s = 0; s < 12; ++s) {
            // B (4x16, mirrored K split): lanes 0-15 -> rows {4s,4s+1},
            // lanes 16-31 -> rows {4s+2,4s+3}; column N = ln.
            const int rb = 16 * i + 4 * s + 2 * hi;
            const int n  = 16 * j + ln;
            v2f B;
            B.x = (rb     < LT) ? sH[(rb)     * TILE + n] : 0.0f;
            B.y = (rb + 1 < LT) ? sH[(rb + 1) * TILE + n] : 0.0f;

            cacc = __builtin_amdgcn_wmma_f32_16x16x4_f32(
                false, Aband[s], false, B, (short)0, cacc, false, false);
        }

        // ---- Stage 4: fused weight + partial reductions ----
        // C/D layout: VGPR e -> M = e + 8*hi, N = ln.
        const int ry0 = 16 * i + 8 * hi;            // row of element e=0
        const int cx  = 16 * j + ln;
        const float* prow = inp + base + (size_t)(y0 + ry0) * IMG_W + (x0 + cx);
        const float* trow = &sT[(ry0 + HALO) * LTS + (cx + HALO)];
        #pragma unroll
        for (int e = 0; e < 8; ++e) {
            const float t  = trow[e * LTS];          // ds_load, imm offsets
            const float in = prow[(size_t)e * IMG_W];// global_load, imm offsets
            const float w  = 1.0f + GAMMA * fabsf(cacc[e] * INV_KK - t);
            const float iw = in * w;
            a0 += iw * t;
            a1 += iw;
            a2 += t * w;
        }
    }

    // ---- Stage 5: wave32 shuffle reduce -> LDS -> three f64 atomics per block ----
    for (int off = 16; off > 0; off >>= 1) {
        a0 += __shfl_down(a0, off, 32);
        a1 += __shfl_down(a1, off, 32);
        a2 += __shfl_down(a2, off, 32);
    }
    if (lane == 0) {
        sRed[wave * 3 + 0] = a0;
        sRed[wave * 3 + 1] = a1;
        sRed[wave * 3 + 2] = a2;
    }
    __syncthreads();
    if (tid == 0) {
        double s0 = 0.0, s1 = 0.0, s2 = 0.0;
        for (int w = 0; w < 8; ++w) {
            s0 += (double)sRed[w * 3 + 0];
            s1 += (double)sRed[w * 3 + 1];
            s2 += (double)sRed[w * 3 + 2];
        }
        atomicAdd(&acc[0], s0);
        atomicAdd(&acc[1], s1);
        atomicAdd(&acc[2], s2);
    }
}

__global__ void wdice_final(const double* __restrict__ acc, float* __restrict__ out) {
    const double inter = acc[0];
    const double denom = acc[1] + acc[2];
    const double dice  = (2.0 * inter + 1.0) / (denom + 1.0);
    out[0] = (float)(1.0 - dice);
}

extern "C" void kernel_launch(void* const* d_in, const int* in_sizes, int n_in,
                              void* d_out, int out_size, void* d_ws, size_t ws_size,
                              hipStream_t stream) {
    (void)in_sizes; (void)n_in; (void)out_size; (void)ws_size;
    const float* inp = (const float*)d_in[0];
    const float* tgt = (const float*)d_in[1];
    double* acc = (double*)d_ws;           // 3 doubles of scratch
    float* out = (float*)d_out;

    wdice_init<<<1, 32, 0, stream>>>(acc);
    dim3 grid(IMG_W / TILE, IMG_H / TILE, 64);
    wdice_main<<<grid, 256, 0, stream>>>(inp, tgt, acc);
    wdice_final<<<1, 1, 0, stream>>>(acc, out);
}